// NAMTuringDecoder2_5076651344122
// MI455X (gfx1250) — compile-verified
//
#include <hip/hip_runtime.h>

// ---------------- constants (match reference) ----------------
#define DIMC    256
#define NT      4
#define HEADC   64
#define HTC     256      // NT*HEADC
#define BATCH   128
#define SEQLEN  96
#define TINL    64
#define TOUTL   64
#define XK      768      // DIM (tgt) + HT (rval) + DIM (h)
#define GATESN  1024
#define QKVAN   1068     // 4*HT + 11*NT
#define QKVASTR 1088     // padded to 64-wide N tiles (17*64)
#define LSTRIDE (BATCH*HTC)   // 32768 floats: l-stride in (l,b,t,c) tensors

typedef float v2f __attribute__((ext_vector_type(2)));
typedef float v8f __attribute__((ext_vector_type(8)));

#define WMMA_F32(a, b, c) \
  __builtin_amdgcn_wmma_f32_16x16x4_f32(false, (a), false, (b), (short)0, (c), false, false)

// ---------------- 16x64-strip f32 WMMA GEMM (compile-time strides) -------
// C[m,n] = sum_k A[m,k]*W[n,k] + bias[n]
// One wave computes a 16(M) x 64(N) strip. All strides/K are template
// constants so every load is base + immediate offset -> clause-batched
// loads with partial s_wait_loadcnt pipelining.
// grid.x = N/64 strips, grid.y = M/16 tiles, block = 1 wave (32 lanes)
template <int LDA, int LDW, int LDC, int K>
__global__ __launch_bounds__(32) void gemm16x64_wmma_f32(
    const float* __restrict__ A, const float* __restrict__ W,
    const float* __restrict__ bias, float* __restrict__ C)
{
  const int lane = threadIdx.x & 31;
  const int m0 = blockIdx.y << 4;
  const int n0 = blockIdx.x << 6;
  const int r  = lane & 15;
  const int k2 = (lane >> 4) << 1;          // 0 or 2: K-subcolumn per half-wave
  const float* ap = A + (size_t)(m0 + r) * LDA + k2;
  const float* wp = W + (size_t)(n0 + r) * LDW + k2;
  v8f acc[4] = {};
#pragma unroll 2
  for (int k = 0; k < K; k += 8) {
    // load 2 A fragments + 4x2 B fragments (constant offsets off two bases)
    v2f a0 = *(const v2f*)(ap + k);
    v2f a1 = *(const v2f*)(ap + k + 4);
    v2f b0[4], b1[4];
#pragma unroll
    for (int j = 0; j < 4; ++j) {
      b0[j] = *(const v2f*)(wp + j * 16 * LDW + k);
      b1[j] = *(const v2f*)(wp + j * 16 * LDW + k + 4);
    }
#pragma unroll
    for (int j = 0; j < 4; ++j) acc[j] = WMMA_F32(a0, b0[j], acc[j]);
#pragma unroll
    for (int j = 0; j < 4; ++j) acc[j] = WMMA_F32(a1, b1[j], acc[j]);
  }
  const int rbase = m0 + ((lane >> 4) << 3);  // lanes 16-31 hold rows M0+8..15
#pragma unroll
  for (int j = 0; j < 4; ++j) {
    const int col = n0 + j * 16 + r;
    const float bv = bias[col];
#pragma unroll
    for (int i = 0; i < 8; ++i)
      C[(size_t)(rbase + i) * LDC + col] = acc[j][i] + bv;
  }
}

// ---------------- init / prep kernels ----------------
__global__ void k_zero(float* __restrict__ p, int n) {
  int i = blockIdx.x * blockDim.x + threadIdx.x;
  if (i < n) p[i] = 0.0f;
}
__global__ void k_pos_init(float* __restrict__ rpos, float* __restrict__ wpos) {
  int i = blockIdx.x * blockDim.x + threadIdx.x;   // (l,B,NT) flat, l==0 <=> i<512
  if (i < TINL * BATCH * NT) {
    float v = (i < BATCH * NT) ? 1.0f : 0.0f;
    rpos[i] = v; wpos[i] = v;
  }
}
__global__ void k_prep_wc(const float* __restrict__ Wih, const float* __restrict__ Whh,
                          float* __restrict__ Wc) {
  int i = blockIdx.x * blockDim.x + threadIdx.x;   // 1024*768
  if (i < GATESN * XK) {
    int j = i / XK, k = i % XK;
    Wc[i] = (k < DIMC + HTC) ? Wih[j * (DIMC + HTC) + k]
                             : Whh[j * DIMC + (k - (DIMC + HTC))];
  }
}
__global__ void k_prep_bc(const float* __restrict__ bih, const float* __restrict__ bhh,
                          float* __restrict__ bc) {
  int i = blockIdx.x * blockDim.x + threadIdx.x;
  if (i < GATESN) bc[i] = bih[i] + bhh[i];
}
__global__ void k_prep_wq(const float* __restrict__ Wq, const float* __restrict__ bq,
                          float* __restrict__ Wqp, float* __restrict__ bqp) {
  int i = blockIdx.x * blockDim.x + threadIdx.x;   // 1088*256
  if (i < QKVASTR * DIMC) {
    int j = i / DIMC, k = i % DIMC;
    Wqp[i] = (j < QKVAN) ? Wq[j * DIMC + k] : 0.0f;
    if (k == 0) bqp[j] = (j < QKVAN) ? bq[j] : 0.0f;
  }
}

// ---------------- per-step: build x = [tgt_t | rval | h] ----------------
__global__ void k_buildx(const float* __restrict__ tgt_t, const float* __restrict__ src,
                         const float* __restrict__ rpos, const float* __restrict__ h,
                         float* __restrict__ x)
{
  int i = blockIdx.x * blockDim.x + threadIdx.x;   // 128*768
  if (i >= BATCH * XK) return;
  int b = i / XK, col = i % XK;
  float v;
  if (col < DIMC) {
    v = tgt_t[b * DIMC + col];
  } else if (col < DIMC + HTC) {
    int cc = col - DIMC;
    int t = cc >> 6;
    float s = 0.0f;
#pragma unroll 4
    for (int l = 0; l < TINL; ++l)
      s += src[(size_t)l * LSTRIDE + b * HTC + cc] * rpos[l * (BATCH * NT) + b * NT + t];
    v = s;
  } else {
    v = h[b * DIMC + (col - DIMC - HTC)];
  }
  x[(size_t)b * XK + col] = v;
}

// ---------------- per-step: LSTM pointwise ----------------
__global__ void k_lstm(const float* __restrict__ gates,
                       float* __restrict__ h, float* __restrict__ c)
{
  int i = blockIdx.x * blockDim.x + threadIdx.x;   // 128*256
  if (i >= BATCH * DIMC) return;
  int b = i >> 8, d = i & 255;
  const float* g = gates + (size_t)b * GATESN;
  float ig = g[d], fg = g[DIMC + d], gg = g[2 * DIMC + d], og = g[3 * DIMC + d];
  float si = 1.0f / (1.0f + __expf(-ig));
  float sf = 1.0f / (1.0f + __expf(-fg));
  float so = 1.0f / (1.0f + __expf(-og));
  float tg = tanhf(gg);
  float cn = sf * c[i] + si * tg;
  float hn = so * tanhf(cn);
  c[i] = cn;
  h[i] = hn;
}

// ---------------- per-step: pointer machine / tape update ----------------
__device__ __forceinline__ float group64_sum(float v, volatile float* s_red,
                                             int tid, int lane64, int grp) {
  s_red[tid] = v; __syncthreads();
#pragma unroll
  for (int off = 32; off > 0; off >>= 1) {
    if (lane64 < off) s_red[tid] += s_red[tid + off];
    __syncthreads();
  }
  float r = s_red[grp << 6];
  __syncthreads();
  return r;
}

__global__ __launch_bounds__(256) void k_update(
    const float* __restrict__ qkva,         // (B, 1088)
    const float* __restrict__ keytape_in,   // (l,b,t,c)
    float* __restrict__ tape_out,           // (l,b,t,c) == d_out
    float* __restrict__ keytape_out,        // (l,b,t,c)
    float* __restrict__ rpos, float* __restrict__ wpos)  // (l,B,NT)
{
  __shared__ float s_qin[256], s_qout[256], s_red[256];
  __shared__ float s_wpos[256], s_rpos[256];
  __shared__ float s_jin[256], s_jout[256];
  __shared__ float s_dir[NT * 8];           // [t*8 + {0..3 read, 4..7 write}]

  const int b   = blockIdx.x;
  const int tid = threadIdx.x;
  const int t   = tid >> 6;                 // tape id for (t,c) phases
  const int c   = tid & 63;                 // channel / lane-in-group

  // load head positions: thread -> (l=tid>>2, tp=tid&3)
  {
    int l = tid >> 2, tp = tid & 3;
    s_wpos[tp * 64 + l] = wpos[l * (BATCH * NT) + b * NT + tp];
    s_rpos[tp * 64 + l] = rpos[l * (BATCH * NT) + b * NT + tp];
  }

  const float* q = qkva + (size_t)b * QKVASTR;
  float qi = q[t * 64 + c];
  float qo = q[HTC + t * 64 + c];
  float ky = q[2 * HTC + t * 64 + c];
  float vl = q[3 * HTC + t * 64 + c];

  // l2-normalize q_in, q_out, key over channel axis (per tape)
  float n;
  n = group64_sum(qi * qi, s_red, tid, c, t);  qi /= fmaxf(sqrtf(n), 1e-12f);
  n = group64_sum(qo * qo, s_red, tid, c, t);  qo /= fmaxf(sqrtf(n), 1e-12f);
  n = group64_sum(ky * ky, s_red, tid, c, t);  ky /= fmaxf(sqrtf(n), 1e-12f);
  s_qin[tid]  = qi;
  s_qout[tid] = qo;

  // softmax of read/write direction (4 values each) per tape
  if (tid < 8) {
    int tp = tid >> 1, w = tid & 1;
    const float* a = q + 4 * HTC + tp * 11 + w * 4;
    float a0 = a[0], a1 = a[1], a2 = a[2], a3 = a[3];
    float m = fmaxf(fmaxf(a0, a1), fmaxf(a2, a3));
    float e0 = __expf(a0 - m), e1 = __expf(a1 - m);
    float e2 = __expf(a2 - m), e3 = __expf(a3 - m);
    float inv = 1.0f / (e0 + e1 + e2 + e3);
    s_dir[tp * 8 + w * 4 + 0] = e0 * inv;
    s_dir[tp * 8 + w * 4 + 1] = e1 * inv;
    s_dir[tp * 8 + w * 4 + 2] = e2 * inv;
    s_dir[tp * 8 + w * 4 + 3] = e3 * inv;
  }
  __syncthreads();

  // oldval / oldkey at current write position (per tape,channel)
  const size_t base = (size_t)b * HTC + t * 64 + c;
  float oldv = 0.0f, oldk = 0.0f;
#pragma unroll 4
  for (int l = 0; l < TOUTL; ++l) {
    float wpv = s_wpos[t * 64 + l];
    oldv += tape_out[(size_t)l * LSTRIDE + base] * wpv;
    oldk += keytape_out[(size_t)l * LSTRIDE + base] * wpv;
  }

  // content-addressing dots (per tape t, slot l=c) using OLD keytape_out
  float din = 0.0f, dout = 0.0f;
  {
    const size_t kb = (size_t)c * LSTRIDE + (size_t)b * HTC + t * 64;  // (l=c, b, t, :)
#pragma unroll 4
    for (int cc = 0; cc < HEADC; cc += 4) {
      float4 ki = *(const float4*)(keytape_in + kb + cc);
      float4 ko = *(const float4*)(keytape_out + kb + cc);
      din  += ki.x * s_qin[t * 64 + cc]     + ki.y * s_qin[t * 64 + cc + 1]
            + ki.z * s_qin[t * 64 + cc + 2] + ki.w * s_qin[t * 64 + cc + 3];
      dout += ko.x * s_qout[t * 64 + cc]     + ko.y * s_qout[t * 64 + cc + 1]
            + ko.z * s_qout[t * 64 + cc + 2] + ko.w * s_qout[t * 64 + cc + 3];
    }
  }
  // normalize jpos over slot axis l (group of 64 = same tape); barriers inside
  // group64_sum also fence the keytape_out reads above from the writes below.
  n = group64_sum(din * din, s_red, tid, c, t);
  s_jin[tid] = din / fmaxf(sqrtf(n), 1e-12f);
  n = group64_sum(dout * dout, s_red, tid, c, t);
  s_jout[tid] = dout / fmaxf(sqrtf(n), 1e-12f);
  __syncthreads();

  // write value/key to tapes: tape += wpos * (value*wd1 - old*wd2)
  {
    float wd1 = s_dir[t * 8 + 4 + 1];
    float wd2 = s_dir[t * 8 + 4 + 2];
    float dv = vl * wd1 - oldv * wd2;
    float dk = ky * wd1 - oldk * wd2;
#pragma unroll 4
    for (int l = 0; l < TOUTL; ++l) {
      float wpv = s_wpos[t * 64 + l];
      tape_out[(size_t)l * LSTRIDE + base]    += wpv * dv;
      keytape_out[(size_t)l * LSTRIDE + base] += wpv * dk;
    }
  }

  // head movement (circular) : thread -> (tape t, slot l=c)
  {
    int lp = (c + 1) & 63, lm = (c + 63) & 63;
    float rd0 = s_dir[t * 8 + 0], rd1 = s_dir[t * 8 + 1];
    float rd2 = s_dir[t * 8 + 2], rd3 = s_dir[t * 8 + 3];
    float wd0 = s_dir[t * 8 + 4], wd1 = s_dir[t * 8 + 5];
    float wd2 = s_dir[t * 8 + 6], wd3 = s_dir[t * 8 + 7];
    float nw = s_wpos[t * 64 + lp] * wd0 + s_wpos[t * 64 + c] * wd1
             + s_wpos[t * 64 + lm] * wd2 + s_jout[tid] * wd3;
    float nr = s_rpos[t * 64 + lp] * rd0 + s_rpos[t * 64 + c] * rd1
             + s_rpos[t * 64 + lm] * rd2 + s_jin[tid] * rd3;
    wpos[c * (BATCH * NT) + b * NT + t] = nw;
    rpos[c * (BATCH * NT) + b * NT + t] = nr;
  }
}

// ---------------- host-side orchestration ----------------
static inline dim3 grid1(int n, int blk) { return dim3((n + blk - 1) / blk); }

extern "C" void kernel_launch(void* const* d_in, const int* in_sizes, int n_in,
                              void* d_out, int out_size, void* d_ws, size_t ws_size,
                              hipStream_t stream) {
  const float* tgt  = (const float*)d_in[0];   // (96,128,256)
  const float* src  = (const float*)d_in[1];   // (64,128,256)
  const float* Wih  = (const float*)d_in[2];   // (1024,512)
  const float* Whh  = (const float*)d_in[3];   // (1024,256)
  const float* bih  = (const float*)d_in[4];
  const float* bhh  = (const float*)d_in[5];
  const float* Wkw  = (const float*)d_in[6];   // (256,256)
  const float* Wkb  = (const float*)d_in[7];
  const float* Wqw  = (const float*)d_in[8];   // (1068,256)
  const float* Wqb  = (const float*)d_in[9];
  (void)in_sizes; (void)n_in; (void)out_size; (void)ws_size;

  float* ws = (float*)d_ws;
  size_t o = 0;
  float* ktin  = ws + o; o += (size_t)TINL * BATCH * HTC;   // 2097152
  float* ktout = ws + o; o += (size_t)TOUTL * BATCH * HTC;  // 2097152
  float* h     = ws + o; o += (size_t)BATCH * DIMC;         // 32768
  float* cst   = ws + o; o += (size_t)BATCH * DIMC;         // 32768 (contiguous w/ h)
  float* rp    = ws + o; o += (size_t)TINL * BATCH * NT;    // 32768
  float* wpz   = ws + o; o += (size_t)TOUTL * BATCH * NT;   // 32768
  float* xbuf  = ws + o; o += (size_t)BATCH * XK;           // 98304
  float* gates = ws + o; o += (size_t)BATCH * GATESN;       // 131072
  float* qkva  = ws + o; o += (size_t)BATCH * QKVASTR;      // 139264
  float* Wc    = ws + o; o += (size_t)GATESN * XK;          // 786432
  float* bc    = ws + o; o += (size_t)GATESN;               // 1024
  float* Wqp   = ws + o; o += (size_t)QKVASTR * DIMC;       // 278528
  float* bqp   = ws + o; o += (size_t)QKVASTR;              // 1088

  float* tape_out = (float*)d_out;                          // (64,128,256)

  // --- init state (d_out / ws are poisoned by harness) ---
  k_zero<<<grid1(TOUTL * BATCH * HTC, 256), 256, 0, stream>>>(ktout, TOUTL * BATCH * HTC);
  k_zero<<<grid1(2 * BATCH * DIMC, 256), 256, 0, stream>>>(h, 2 * BATCH * DIMC); // h & c
  k_zero<<<grid1(TOUTL * BATCH * HTC, 256), 256, 0, stream>>>(tape_out, TOUTL * BATCH * HTC);
  k_pos_init<<<grid1(TINL * BATCH * NT, 256), 256, 0, stream>>>(rp, wpz);

  // --- prep packed weights ---
  k_prep_wc<<<grid1(GATESN * XK, 256), 256, 0, stream>>>(Wih, Whh, Wc);
  k_prep_bc<<<grid1(GATESN, 256), 256, 0, stream>>>(bih, bhh, bc);
  k_prep_wq<<<grid1(QKVASTR * DIMC, 256), 256, 0, stream>>>(Wqw, Wqb, Wqp, bqp);

  // --- keytape_in = src @ Wk^T + b : (8192 x 256) x (256 x 256) ---
  {
    dim3 g(HTC / 64, (TINL * BATCH) / 16);  // (4, 512)
    gemm16x64_wmma_f32<DIMC, DIMC, HTC, DIMC>
        <<<g, 32, 0, stream>>>(src, Wkw, Wkb, ktin);
  }

  // --- sequential scan over 96 steps ---
  for (int tstep = 0; tstep < SEQLEN; ++tstep) {
    const float* tgt_t = tgt + (size_t)tstep * BATCH * DIMC;

    k_buildx<<<grid1(BATCH * XK, 256), 256, 0, stream>>>(tgt_t, src, rp, h, xbuf);

    {  // gates = x @ Wc^T + bc : (128 x 768) x (768 x 1024)
      dim3 g(GATESN / 64, BATCH / 16);  // (16, 8)
      gemm16x64_wmma_f32<XK, XK, GATESN, XK>
          <<<g, 32, 0, stream>>>(xbuf, Wc, bc, gates);
    }

    k_lstm<<<grid1(BATCH * DIMC, 256), 256, 0, stream>>>(gates, h, cst);

    {  // qkva = h @ Wq^T + bq : (128 x 256) x (256 x 1088)
      dim3 g(QKVASTR / 64, BATCH / 16);  // (17, 8)
      gemm16x64_wmma_f32<DIMC, DIMC, QKVASTR, DIMC>
          <<<g, 32, 0, stream>>>(h, Wqp, bqp, qkva);
    }

    k_update<<<BATCH, 256, 0, stream>>>(qkva, ktin, tape_out, ktout, rp, wpz);
  }
}